// RadonTransform_59760174956954
// MI455X (gfx1250) — compile-verified
//
#include <hip/hip_runtime.h>

#define N_IMG 256
#define LDS_STRIDE 257   // 256 + 1 DWORD pad per row -> bank-conflict-free vertical rays

typedef __attribute__((ext_vector_type(4))) unsigned int uint4v;
typedef __attribute__((ext_vector_type(8))) unsigned int uint8v;

__global__ __launch_bounds__(256, 1)
void radon_tdm_kernel(const float* __restrict__ x,
                      const float* __restrict__ angles,
                      const float* __restrict__ positions,
                      float* __restrict__ out,
                      int A)
{
    extern __shared__ float lds[];   // 257*256 floats = 263,168 B (< 320 KB WGP LDS)

    const int a = blockIdx.x;        // angle index  (grid.x = A)
    const int b = blockIdx.y;        // batch index  (grid.y = B)
    const int p = threadIdx.x;       // detector position, P == 256 == blockDim.x

    // ---- Stage image b (256x256 f32, 256 KB) into LDS with the Tensor Data
    // Mover. Descriptor pads LDS dest by 1 DWORD every 256 DWORDs (row stride
    // 257). Issued by wave 0 only (TDM is a whole-wave DMA, EXEC ignored).
    if (threadIdx.x < 32) {
        unsigned long long ga =
            (unsigned long long)(const void*)(x + (size_t)b * N_IMG * N_IMG);

        uint4v g0;
        g0[0] = 1u;                                      // count=1, user D#, no gather
        g0[1] = 0u;                                      // lds_addr = 0
        g0[2] = (unsigned)(ga & 0xFFFFFFFFu);            // global_addr[31:0]
        g0[3] = (unsigned)((ga >> 32) & 0x01FFFFFFu)     // global_addr[56:32]
              | (2u << 30);                              // type = 2 ("image")

        uint8v g1;
        g1[0] = (2u << 16)                               // data_size = 4 bytes
              | (1u << 20)                               // pad_enable
              | (7u << 22)                               // pad_interval: 256 DWORDs
              | (0u << 25);                              // pad_amount: 1 DWORD
        g1[1] = ((unsigned)N_IMG) << 16;                 // tensor_dim0[15:0]=256 (bits 79:48)
        g1[2] = ((unsigned)N_IMG) << 16;                 // tensor_dim0 hi=0 | tensor_dim1[15:0]=256
        g1[3] = ((unsigned)N_IMG) << 16;                 // tensor_dim1 hi=0 | tile_dim0=256
        g1[4] = (unsigned)N_IMG;                         // tile_dim1=256, tile_dim2=0
        g1[5] = (unsigned)N_IMG;                         // tensor_dim0_stride = 256 elements
        g1[6] = 0u;                                      // stride hi / tensor_dim1_stride lo
        g1[7] = 0u;

        asm volatile("tensor_load_to_lds %0, %1"
                     :
                     : "s"(g0), "s"(g1)
                     : "memory");
        __builtin_amdgcn_s_wait_tensorcnt(0);
    }
    __syncthreads();

    // ---- Per-thread ray march.
    const float ang = angles[a];
    const float cA  = cosf(ang);
    const float sA  = sinf(ang);
    const float s   = positions[p];
    const float c0  = (N_IMG - 1) * 0.5f;          // 127.5
    const float bx  = fmaf(s, cA, c0);             // px = bx - t*sA
    const float by  = fmaf(s, sA, c0);             // py = by + t*cA

    float acc = 0.0f;
#pragma unroll 8
    for (int t = 0; t < N_IMG; ++t) {
        const float tv = (float)t - (N_IMG - 1) * 0.5f;
        const float px = fmaf(-sA, tv, bx);
        const float py = fmaf( cA, tv, by);
        const float fx = floorf(px);
        const float fy = floorf(py);
        const int   ix = (int)fx;
        const int   iy = (int)fy;
        const float wx = px - fx;
        const float wy = py - fy;
        const int   ix1 = ix + 1;
        const int   iy1 = iy + 1;

        // per-corner validity folded into the separable weights (matches ref: w*valid)
        const float ax0 = ((unsigned)ix  < (unsigned)N_IMG) ? (1.0f - wx) : 0.0f;
        const float ax1 = ((unsigned)ix1 < (unsigned)N_IMG) ? wx          : 0.0f;
        const float ay0 = ((unsigned)iy  < (unsigned)N_IMG) ? (1.0f - wy) : 0.0f;
        const float ay1 = ((unsigned)iy1 < (unsigned)N_IMG) ? wy          : 0.0f;

        const int x0c = min(max(ix,  0), N_IMG - 1);
        const int x1c = min(max(ix1, 0), N_IMG - 1);
        const int y0c = min(max(iy,  0), N_IMG - 1);
        const int y1c = min(max(iy1, 0), N_IMG - 1);

        const float* r0 = lds + y0c * LDS_STRIDE;
        const float* r1 = lds + y1c * LDS_STRIDE;

        acc = fmaf(ax0 * ay0, r0[x0c], acc);
        acc = fmaf(ax1 * ay0, r0[x1c], acc);
        acc = fmaf(ax0 * ay1, r1[x0c], acc);
        acc = fmaf(ax1 * ay1, r1[x1c], acc);
    }

    // out[b, 0, a, p] with C==1, P==256: flat index (b*A + a)*256 + p
    out[((size_t)b * A + a) * N_IMG + p] = acc;
}

extern "C" void kernel_launch(void* const* d_in, const int* in_sizes, int n_in,
                              void* d_out, int out_size, void* d_ws, size_t ws_size,
                              hipStream_t stream) {
    const float* x         = (const float*)d_in[0];
    const float* angles    = (const float*)d_in[1];
    const float* positions = (const float*)d_in[2];
    float* out             = (float*)d_out;

    const int A = in_sizes[1];                       // 180
    const int B = in_sizes[0] / (N_IMG * N_IMG);     // 2

    const size_t ldsBytes = (size_t)LDS_STRIDE * N_IMG * sizeof(float);
    radon_tdm_kernel<<<dim3(A, B), dim3(N_IMG), ldsBytes, stream>>>(
        x, angles, positions, out, A);
}